// Whole_Encoder_21492016349191
// MI455X (gfx1250) — compile-verified
//
#include <hip/hip_runtime.h>
#include <hip/hip_bf16.h>
#include <math.h>

// ---------------- problem constants ----------------
#define S_    128
#define L_    128
#define E_    300     // embedding dim (divisible by 4 -> clean K-loop for 16x16x4)
#define H_    300     // hidden per direction
#define G4_   1200    // 4*H
#define D_    600     // 2*H
#define DP_   608     // D padded to multiple of 16 for WMMA N-tiles
#define NMAX_ (S_*L_) // 16384
#define NEGV  (-1e30f)

typedef float v2f __attribute__((ext_vector_type(2)));
typedef float v8f __attribute__((ext_vector_type(8)));

// ---- optional gfx1250 async global->LDS path (guarded; falls back cleanly) ----
#if defined(__has_builtin)
#  if __has_builtin(__builtin_amdgcn_global_load_async_to_lds_b128) && \
      __has_builtin(__builtin_amdgcn_s_wait_asynccnt)
#    define HAVE_ASYNC_LDS 1
#  endif
#endif
#ifndef HAVE_ASYNC_LDS
#  define HAVE_ASYNC_LDS 0
#endif

#if HAVE_ASYNC_LDS
typedef int v4i_ __attribute__((vector_size(4 * sizeof(int))));
typedef __attribute__((address_space(1))) v4i_ as1_v4i;   // global
typedef __attribute__((address_space(3))) v4i_ as3_v4i;   // LDS
__device__ __forceinline__ void cp16_async(float* lds_dst, const float* g_src) {
    __builtin_amdgcn_global_load_async_to_lds_b128(
        (as1_v4i*)g_src, (as3_v4i*)lds_dst, 0, 0);
}
__device__ __forceinline__ void cp_async_wait() {
    __builtin_amdgcn_s_wait_asynccnt(0);
}
#else
__device__ __forceinline__ void cp16_async(float* lds_dst, const float* g_src) {
    *(float4*)lds_dst = *(const float4*)g_src;   // synchronous fallback
}
__device__ __forceinline__ void cp_async_wait() {}
#endif

// ---------------------------------------------------------------------------
// Kernel 1: prefix-sum of lengths + flat (sentence,pos,token-id) map.
// ---------------------------------------------------------------------------
__global__ void k_build_map(const int* __restrict__ tokens,
                            const int* __restrict__ lengths,
                            int* __restrict__ Nptr,
                            int* __restrict__ rows,
                            int* __restrict__ cols,
                            int* __restrict__ fids) {
    __shared__ int offs[S_];
    const int tid = threadIdx.x;
    if (tid == 0) {
        int run = 0;
        for (int s = 0; s < S_; ++s) { offs[s] = run; run += lengths[s]; }
        *Nptr = run;
    }
    __syncthreads();
    const int s    = tid;
    const int base = offs[s];
    const int len  = lengths[s];
    for (int p = 0; p < len; ++p) {
        const int n = base + p;
        rows[n] = s;
        cols[n] = p;
        fids[n] = tokens[s * L_ + p];
    }
}

// ---------------------------------------------------------------------------
// Kernel 2: embedding gather; rows >= N zero-filled for clean WMMA tiles.
// ---------------------------------------------------------------------------
__global__ void k_gather(const float* __restrict__ table,
                         const int* __restrict__ fids,
                         const int* __restrict__ Nptr,
                         float* __restrict__ embeds) {
    const int n = blockIdx.x;
    const int t = threadIdx.x;
    if (t >= E_ / 4) return;
    float4 v = make_float4(0.f, 0.f, 0.f, 0.f);
    if (n < *Nptr) {
        const float4* src = (const float4*)(table + (size_t)fids[n] * E_);
        v = src[t];
    }
    ((float4*)(embeds + (size_t)n * E_))[t] = v;
}

// ---------------------------------------------------------------------------
// Kernel 3: xw = embeds @ Wih^T + (bih+bhh), both dirs.  One wave computes a
// 16x48 strip (3 accumulators) so the A fragment is reused 3x per k-step.
// ---------------------------------------------------------------------------
__global__ void k_proj(const float* __restrict__ embeds,
                       const float* __restrict__ Wih_f,
                       const float* __restrict__ bih_f,
                       const float* __restrict__ bhh_f,
                       const float* __restrict__ Wih_b,
                       const float* __restrict__ bih_b,
                       const float* __restrict__ bhh_b,
                       const int* __restrict__ Nptr,
                       float* __restrict__ xw_f,
                       float* __restrict__ xw_b) {
    const int row0 = blockIdx.x * 16;
    const int j0   = blockIdx.y * 48;
    const int dir  = blockIdx.z;
    const int N    = *Nptr;
    if (row0 >= N) return;                 // block-uniform -> EXEC stays all-1s

    const float* W   = dir ? Wih_b : Wih_f;
    const float* b1  = dir ? bih_b : bih_f;
    const float* b2  = dir ? bhh_b : bhh_f;
    float*       out = dir ? xw_b  : xw_f;

    const int lane = threadIdx.x & 31;
    const int lo   = lane & 15;
    const int hi   = lane >> 4;
    const int kk   = hi * 2;

    v8f c0 = {}, c1 = {}, c2 = {};
    const float* Arow = embeds + (size_t)(row0 + lo) * E_;
    const float* W0   = W + (size_t)(j0      + lo) * E_;
    const float* W1   = W + (size_t)(j0 + 16 + lo) * E_;
    const float* W2   = W + (size_t)(j0 + 32 + lo) * E_;
    for (int k0 = 0; k0 < E_; k0 += 4) {
        v2f a, b0, b1v, b2v;
        a.x   = Arow[k0 + kk];  a.y   = Arow[k0 + kk + 1];
        b0.x  = W0[k0 + kk];    b0.y  = W0[k0 + kk + 1];
        b1v.x = W1[k0 + kk];    b1v.y = W1[k0 + kk + 1];
        b2v.x = W2[k0 + kk];    b2v.y = W2[k0 + kk + 1];
        c0 = __builtin_amdgcn_wmma_f32_16x16x4_f32(false, a, false, b0,
                                                   (short)0, c0, false, false);
        c1 = __builtin_amdgcn_wmma_f32_16x16x4_f32(false, a, false, b1v,
                                                   (short)0, c1, false, false);
        c2 = __builtin_amdgcn_wmma_f32_16x16x4_f32(false, a, false, b2v,
                                                   (short)0, c2, false, false);
    }
    const float bias0 = b1[j0      + lo] + b2[j0      + lo];
    const float bias1 = b1[j0 + 16 + lo] + b2[j0 + 16 + lo];
    const float bias2 = b1[j0 + 32 + lo] + b2[j0 + 32 + lo];
    #pragma unroll
    for (int r = 0; r < 8; ++r) {
        const int m = r + 8 * hi;
        float* orow = out + (size_t)(row0 + m) * G4_;
        orow[j0      + lo] = c0[r] + bias0;
        orow[j0 + 16 + lo] = c1[r] + bias1;
        orow[j0 + 32 + lo] = c2[r] + bias2;
    }
}

// ---------------------------------------------------------------------------
// Kernel 4: zero the padded [S,L,DP] buffer (ws is poisoned by the harness).
// ---------------------------------------------------------------------------
__global__ void k_zero(float4* __restrict__ p, int n4) {
    const int i = blockIdx.x * blockDim.x + threadIdx.x;
    if (i < n4) p[i] = make_float4(0.f, 0.f, 0.f, 0.f);
}

// ---------------------------------------------------------------------------
// Kernel 5: sequential LSTM scan, one block per direction.  h/c in LDS,
// Whh streamed from L2.  Next timestep's 4.8KB gate row double-buffered into
// LDS with gfx1250 async global->LDS loads, retired with s_wait_asynccnt.
// ---------------------------------------------------------------------------
__global__ void k_lstm(const float* __restrict__ Whh_f,
                       const float* __restrict__ Whh_b,
                       const float* __restrict__ xw_f,
                       const float* __restrict__ xw_b,
                       const int* __restrict__ Nptr,
                       const int* __restrict__ rows,
                       const int* __restrict__ cols,
                       float* __restrict__ padded) {
    const int dir = blockIdx.x;
    const float* Whh = dir ? Whh_b : Whh_f;
    const float* xw  = dir ? xw_b  : xw_f;

    __shared__ __align__(16) float hs[H_];
    __shared__ __align__(16) float cs[H_];
    __shared__ __align__(16) float xbuf[2][G4_];
    const int j = threadIdx.x;
    if (j < H_) { hs[j] = 0.f; cs[j] = 0.f; }

    const int N = *Nptr;
    // preload gate row for the first timestep
    if (N > 0 && j < H_) {
        const int t0 = dir ? (N - 1) : 0;
        cp16_async(&xbuf[0][4 * j], xw + (size_t)t0 * G4_ + 4 * j);
    }
    cp_async_wait();
    __syncthreads();

    for (int t = 0; t < N; ++t) {
        const int cur = t & 1, nxt = cur ^ 1;
        const int ti  = dir ? (N - 1 - t) : t;

        // kick off the async copy of the next timestep's gate row
        if (t + 1 < N && j < H_) {
            const int tn = dir ? (N - 2 - t) : (t + 1);
            cp16_async(&xbuf[nxt][4 * j], xw + (size_t)tn * G4_ + 4 * j);
        }

        float hn = 0.f, cn = 0.f;
        if (j < H_) {
            const float* xr = xbuf[cur];
            float g[4];
            #pragma unroll
            for (int k = 0; k < 4; ++k) {
                const float4* wr = (const float4*)(Whh + (size_t)(k * H_ + j) * H_);
                const float4* hv = (const float4*)hs;
                float acc = 0.f;
                for (int m = 0; m < H_ / 4; ++m) {
                    const float4 w  = wr[m];
                    const float4 h4 = hv[m];
                    acc = fmaf(w.x, h4.x, fmaf(w.y, h4.y,
                          fmaf(w.z, h4.z, fmaf(w.w, h4.w, acc))));
                }
                g[k] = xr[k * H_ + j] + acc;
            }
            const float ig = 1.f / (1.f + expf(-g[0]));
            const float fg = 1.f / (1.f + expf(-g[1]));
            const float gg = tanhf(g[2]);
            const float og = 1.f / (1.f + expf(-g[3]));
            cn = fg * cs[j] + ig * gg;
            hn = og * tanhf(cn);
        }
        __syncthreads();                 // all reads of hs/cs for step t done
        if (j < H_) {
            hs[j] = hn;
            cs[j] = cn;
            const int r = rows[ti], p = cols[ti];
            padded[(size_t)(r * L_ + p) * DP_ + dir * H_ + j] = hn;
        }
        cp_async_wait();                 // next row is fully in LDS
        __syncthreads();                 // publish h_t and xbuf[nxt]
    }
}

// ---------------------------------------------------------------------------
// Kernel 6: scores = (P P^T)/sqrt(D) with key mask.  One wave computes a
// 16x64 strip (4 key tiles) so the A fragment is reused 4x per k-step.
// ---------------------------------------------------------------------------
__global__ void k_scores(const float* __restrict__ padded,
                         const int* __restrict__ lengths,
                         float* __restrict__ scores) {
    const int q0 = blockIdx.x * 16;
    const int k0 = blockIdx.y * 64;
    const int s  = blockIdx.z;
    const float* P = padded + (size_t)s * L_ * DP_;

    const int lane = threadIdx.x & 31;
    const int lo   = lane & 15;
    const int hi   = lane >> 4;
    const int kk   = hi * 2;

    v8f c[4] = {{}, {}, {}, {}};
    const float* Arow = P + (size_t)(q0 + lo) * DP_;
    const float* Brow[4];
    #pragma unroll
    for (int tI = 0; tI < 4; ++tI)
        Brow[tI] = P + (size_t)(k0 + 16 * tI + lo) * DP_;   // B^T rows

    for (int kd = 0; kd < DP_; kd += 4) {
        v2f a;
        a.x = Arow[kd + kk];  a.y = Arow[kd + kk + 1];
        #pragma unroll
        for (int tI = 0; tI < 4; ++tI) {
            v2f b;
            b.x = Brow[tI][kd + kk];  b.y = Brow[tI][kd + kk + 1];
            c[tI] = __builtin_amdgcn_wmma_f32_16x16x4_f32(false, a, false, b,
                                                          (short)0, c[tI],
                                                          false, false);
        }
    }
    const int   len   = lengths[s];
    const float scale = rsqrtf((float)D_);
    float* out = scores + (size_t)s * L_ * L_;
    #pragma unroll
    for (int tI = 0; tI < 4; ++tI) {
        const bool kvalid = (k0 + 16 * tI + lo) < len;
        #pragma unroll
        for (int r = 0; r < 8; ++r) {
            const int m = r + 8 * hi;
            out[(size_t)(q0 + m) * L_ + k0 + 16 * tI + lo] =
                kvalid ? c[tI][r] * scale : NEGV;
        }
    }
}

// ---------------------------------------------------------------------------
// Kernel 7: row softmax over 128 keys, in place (matches jax on fully-masked
// rows: uniform 1/128).
// ---------------------------------------------------------------------------
__global__ void k_softmax(float* __restrict__ scores) {
    const int row = blockIdx.x;       // S*L rows
    const int tid = threadIdx.x;      // 128
    __shared__ float red[L_];
    float* x = scores + (size_t)row * L_;

    const float v = x[tid];
    red[tid] = v; __syncthreads();
    for (int o = 64; o > 0; o >>= 1) {
        if (tid < o) red[tid] = fmaxf(red[tid], red[tid + o]);
        __syncthreads();
    }
    const float mx = red[0];
    __syncthreads();
    const float e = expf(v - mx);
    red[tid] = e; __syncthreads();
    for (int o = 64; o > 0; o >>= 1) {
        if (tid < o) red[tid] += red[tid + o];
        __syncthreads();
    }
    x[tid] = e / red[0];
}

// ---------------------------------------------------------------------------
// Kernel 8: ctx = attn @ P fused with masked max-pool over q.  Block = 8
// waves; wave w owns q rows [16w,16w+16) x 32 output cols (2 accumulators),
// then the block max-reduces over all 128 q rows.
// ---------------------------------------------------------------------------
__global__ void k_ctx_maxpool(const float* __restrict__ scores,
                              const float* __restrict__ padded,
                              const int* __restrict__ lengths,
                              float* __restrict__ out) {
    const int s    = blockIdx.x;
    const int d0   = blockIdx.y * 32;
    const int wave = threadIdx.x >> 5;   // 0..7
    const int lane = threadIdx.x & 31;
    const int lo   = lane & 15;
    const int hi   = lane >> 4;
    const int kk   = hi * 2;
    const int q0   = wave * 16;

    const float* A = scores + (size_t)s * L_ * L_;   // attn (post-softmax)
    const float* P = padded + (size_t)s * L_ * DP_;

    v8f c0 = {}, c1 = {};
    const float* Arow = A + (size_t)(q0 + lo) * L_;
    for (int kd = 0; kd < L_; kd += 4) {
        v2f a, b0, b1;
        a.x  = Arow[kd + kk];  a.y  = Arow[kd + kk + 1];
        b0.x = P[(size_t)(kd + kk)     * DP_ + d0 + lo];
        b0.y = P[(size_t)(kd + kk + 1) * DP_ + d0 + lo];
        b1.x = P[(size_t)(kd + kk)     * DP_ + d0 + 16 + lo];
        b1.y = P[(size_t)(kd + kk + 1) * DP_ + d0 + 16 + lo];
        c0 = __builtin_amdgcn_wmma_f32_16x16x4_f32(false, a, false, b0,
                                                   (short)0, c0, false, false);
        c1 = __builtin_amdgcn_wmma_f32_16x16x4_f32(false, a, false, b1,
                                                   (short)0, c1, false, false);
    }
    const int len = lengths[s];
    float mx0 = NEGV, mx1 = NEGV;
    #pragma unroll
    for (int r = 0; r < 8; ++r) {
        const int q = q0 + r + 8 * hi;
        const bool qv = q < len;
        mx0 = fmaxf(mx0, qv ? c0[r] : NEGV);
        mx1 = fmaxf(mx1, qv ? c1[r] : NEGV);
    }
    // lanes l and l+16 hold the same output column -> fold halves
    mx0 = fmaxf(mx0, __shfl_xor(mx0, 16, 32));
    mx1 = fmaxf(mx1, __shfl_xor(mx1, 16, 32));

    __shared__ float red[8][32];
    if (hi == 0) {
        red[wave][lo]      = mx0;
        red[wave][16 + lo] = mx1;
    }
    __syncthreads();
    if (wave == 0) {                     // lanes 0..31 -> 32 output columns
        float m = red[0][lane];
        #pragma unroll
        for (int w = 1; w < 8; ++w) m = fmaxf(m, red[w][lane]);
        const int d = d0 + lane;
        if (d < D_) out[(size_t)s * D_ + d] = m;
    }
}

// ---------------------------------------------------------------------------
extern "C" void kernel_launch(void* const* d_in, const int* in_sizes, int n_in,
                              void* d_out, int out_size, void* d_ws, size_t ws_size,
                              hipStream_t stream) {
    const int*   tokens  = (const int*)  d_in[0];
    const int*   lengths = (const int*)  d_in[1];
    const float* table   = (const float*)d_in[2];
    const float* Wih_f   = (const float*)d_in[3];
    const float* Whh_f   = (const float*)d_in[4];
    const float* bih_f   = (const float*)d_in[5];
    const float* bhh_f   = (const float*)d_in[6];
    const float* Wih_b   = (const float*)d_in[7];
    const float* Whh_b   = (const float*)d_in[8];
    const float* bih_b   = (const float*)d_in[9];
    const float* bhh_b   = (const float*)d_in[10];
    float* out = (float*)d_out;

    // ---- workspace carve-out (256B aligned slices) ----
    char* w = (char*)d_ws;
    auto alloc = [&](size_t bytes) -> char* {
        char* p = w;
        w += (bytes + 255) & ~(size_t)255;
        return p;
    };
    int*   Nptr   = (int*)  alloc(sizeof(int));
    int*   rows   = (int*)  alloc((size_t)NMAX_ * sizeof(int));
    int*   cols   = (int*)  alloc((size_t)NMAX_ * sizeof(int));
    int*   fids   = (int*)  alloc((size_t)NMAX_ * sizeof(int));
    float* embeds = (float*)alloc((size_t)NMAX_ * E_  * sizeof(float));
    float* xw_f   = (float*)alloc((size_t)NMAX_ * G4_ * sizeof(float));
    float* xw_b   = (float*)alloc((size_t)NMAX_ * G4_ * sizeof(float));
    float* padded = (float*)alloc((size_t)S_ * L_ * DP_ * sizeof(float));
    float* scores = (float*)alloc((size_t)S_ * L_ * L_  * sizeof(float));

    // 1. ragged map + total N (device-side; lengths live on device)
    k_build_map<<<1, 128, 0, stream>>>(tokens, lengths, Nptr, rows, cols, fids);

    // 2. embedding gather (zero-fills rows >= N)
    k_gather<<<NMAX_, 96, 0, stream>>>(table, fids, Nptr, embeds);

    // 3. input projections for both directions (fp32 WMMA, 16x48 per wave)
    {
        dim3 g(NMAX_ / 16, G4_ / 48, 2);
        k_proj<<<g, 32, 0, stream>>>(embeds, Wih_f, bih_f, bhh_f,
                                     Wih_b, bih_b, bhh_b, Nptr, xw_f, xw_b);
    }

    // 4. zero padded attention buffer, then the two serial scan chains
    {
        size_t n4 = (size_t)S_ * L_ * DP_ / 4;
        k_zero<<<(unsigned)((n4 + 255) / 256), 256, 0, stream>>>((float4*)padded, (int)n4);
    }
    k_lstm<<<2, 320, 0, stream>>>(Whh_f, Whh_b, xw_f, xw_b, Nptr, rows, cols, padded);

    // 5. attention: scores -> softmax -> ctx + fused masked max-pool
    {
        dim3 g(L_ / 16, L_ / 64, S_);
        k_scores<<<g, 32, 0, stream>>>(padded, lengths, scores);
    }
    k_softmax<<<S_ * L_, L_, 0, stream>>>(scores);
    {
        dim3 g(S_, DP_ / 32);
        k_ctx_maxpool<<<g, 256, 0, stream>>>(scores, padded, lengths, out);
    }
}